// HyperDiffRecCore_13975823581875
// MI455X (gfx1250) — compile-verified
//
#include <hip/hip_runtime.h>
#include <math.h>

#define D 64
#define K2 128
#define M_USERS 200000
#define NUM_NODES 300000

typedef __attribute__((ext_vector_type(2))) float v2f;
typedef __attribute__((ext_vector_type(8))) float v8f;

// ---------------- zero workspace ----------------
__global__ void zero_kernel(float4* __restrict__ p, long n4) {
    long i = (long)blockIdx.x * blockDim.x + threadIdx.x;
    long stride = (long)gridDim.x * blockDim.x;
    float4 z = make_float4(0.f, 0.f, 0.f, 0.f);
    for (; i < n4; i += stride) p[i] = z;
}

// ---------------- SpMM: one wave (32 lanes) per edge ----------------
// out[row] += val * x[col];  x is split into two tensors at `split`
__global__ void spmm_kernel(const float* __restrict__ xu,
                            const float* __restrict__ xi,
                            int split,
                            const int*   __restrict__ rows,
                            const int*   __restrict__ cols,
                            const float* __restrict__ vals,
                            float* __restrict__ out,
                            int nedges) {
    int warp = (blockIdx.x * blockDim.x + threadIdx.x) >> 5;
    int lane = threadIdx.x & 31;
    int nw = (gridDim.x * blockDim.x) >> 5;
    for (int e = warp; e < nedges; e += nw) {
        int r = rows[e];
        int c = cols[e];
        float v = vals[e];
        const float* src = (c < split) ? (xu + (size_t)c * D)
                                       : (xi + (size_t)(c - split) * D);
        float2 xv = *(const float2*)(src + lane * 2);
        float* dst = out + (size_t)r * D + lane * 2;
        atomicAdd(dst,     v * xv.x);
        atomicAdd(dst + 1, v * xv.y);
    }
}

// ---------------- Fusion MLP with f32 WMMA ----------------
// Per wave: 16 nodes. A = [f1|f2] (16x128) staged in LDS, W1^T staged in LDS.
// GEMM 16x64x128 via 32 k-steps x 4 n-tiles of v_wmma_f32_16x16x4_f32.
__global__ __launch_bounds__(128) void fuse_kernel(
    const float* __restrict__ user_emb, const float* __restrict__ item_emb,
    const float* __restrict__ hg_user,  const float* __restrict__ hg_item,
    const float* __restrict__ h1,       const float* __restrict__ h2,
    const float* __restrict__ W1,       const float* __restrict__ b1,
    const float* __restrict__ W2,       const float* __restrict__ b2,
    float* __restrict__ out)
{
    __shared__ float sW1t[D * K2];      // W1 transposed: [n][k], 32 KB
    __shared__ float sA[4][16 * K2];    // per-wave A staging, 32 KB

    const int tid  = threadIdx.x;
    const int w    = tid >> 5;
    const int lane = tid & 31;

    // Stage W1 transposed into LDS (row-major W1 is [k*64+n])
    for (int i = tid; i < K2 * D; i += 128) {
        int k = i >> 6;
        int n = i & 63;
        sW1t[n * K2 + k] = W1[i];
    }

    const int tile_base = (blockIdx.x * 4 + w) * 16;

    // Stage A rows: node-feature matrix [f1(64) | f2(64)], float4 granularity
    for (int i = lane; i < 16 * 32; i += 32) {
        int node = i >> 5;          // 0..15
        int c4   = i & 31;          // float4 index within 128 floats
        int feat = c4 * 4;
        int ng   = tile_base + node;
        float4 val = make_float4(0.f, 0.f, 0.f, 0.f);
        if (ng < NUM_NODES) {
            if (feat < D) {
                const float* ego = (ng < M_USERS)
                    ? (user_emb + (size_t)ng * D + feat)
                    : (item_emb + (size_t)(ng - M_USERS) * D + feat);
                float4 e4 = *(const float4*)ego;
                float4 a4 = *(const float4*)(h1 + (size_t)ng * D + feat);
                float4 b4 = *(const float4*)(h2 + (size_t)ng * D + feat);
                const float inv3 = 1.0f / 3.0f;
                val = make_float4((e4.x + a4.x + b4.x) * inv3,
                                  (e4.y + a4.y + b4.y) * inv3,
                                  (e4.z + a4.z + b4.z) * inv3,
                                  (e4.w + a4.w + b4.w) * inv3);
            } else {
                int f = feat - D;
                const float* hg = (ng < M_USERS)
                    ? (hg_user + (size_t)ng * D + f)
                    : (hg_item + (size_t)(ng - M_USERS) * D + f);
                val = *(const float4*)hg;
            }
        }
        *(float4*)&sA[w][node * K2 + feat] = val;
    }

    __syncthreads();

    // A layout (f32 16x4): lanes 0-15 -> rows M=lane, K = k0,k0+1 ;
    //                      lanes 16-31 -> rows M=lane-16, K = k0+2,k0+3
    const int mrow = lane & 15;
    const int koff = (lane >> 4) * 2;

    v8f acc[4] = {};
    for (int kk = 0; kk < 32; ++kk) {
        int k0 = kk * 4 + koff;
        v2f a;
        a.x = sA[w][mrow * K2 + k0];
        a.y = sA[w][mrow * K2 + k0 + 1];
        #pragma unroll
        for (int nt = 0; nt < 4; ++nt) {
            int ncol = nt * 16 + mrow;
            v2f b;
            b.x = sW1t[ncol * K2 + k0];
            b.y = sW1t[ncol * K2 + k0 + 1];
            acc[nt] = __builtin_amdgcn_wmma_f32_16x16x4_f32(
                false, a, false, b, (short)0, acc[nt], false, false);
        }
    }

    // hdn = tanh(acc + b1[n]); partial logits vs W2 (64x2)
    float p0[8], p1[8];
    #pragma unroll
    for (int v = 0; v < 8; ++v) { p0[v] = 0.f; p1[v] = 0.f; }
    for (int nt = 0; nt < 4; ++nt) {
        int ncol = nt * 16 + mrow;
        float bb  = b1[ncol];
        float w2a = W2[ncol * 2 + 0];
        float w2b = W2[ncol * 2 + 1];
        #pragma unroll
        for (int v = 0; v < 8; ++v) {
            float hdn = tanhf(acc[nt][v] + bb);
            p0[v] += hdn * w2a;
            p1[v] += hdn * w2b;
        }
    }

    // Reduce across the 16 lanes of each half (xor masks stay within halves)
    #pragma unroll
    for (int off = 1; off < 16; off <<= 1) {
        #pragma unroll
        for (int v = 0; v < 8; ++v) {
            p0[v] += __shfl_xor(p0[v], off, 32);
            p1[v] += __shfl_xor(p1[v], off, 32);
        }
    }

    float bb0 = b2[0], bb1 = b2[1];
    const int half = lane >> 4;   // rows v (half==0) or v+8 (half==1)

    for (int v = 0; v < 8; ++v) {
        float l0 = p0[v] + bb0;
        float l1 = p1[v] + bb1;
        float e  = expf(l1 - l0);
        float w1w = e / (1.0f + e);
        float w0w = 1.0f - w1w;
        int m  = v + half * 8;
        int ng = tile_base + m;
        if (ng < NUM_NODES) {
            #pragma unroll
            for (int ff = 0; ff < 4; ++ff) {
                int feat = mrow + ff * 16;
                float f1v = sA[w][m * K2 + feat];
                float f2v = sA[w][m * K2 + D + feat];
                out[(size_t)ng * D + feat] = w0w * f1v + w1w * f2v;
            }
        }
    }
}

extern "C" void kernel_launch(void* const* d_in, const int* in_sizes, int n_in,
                              void* d_out, int out_size, void* d_ws, size_t ws_size,
                              hipStream_t stream) {
    const float* user_emb = (const float*)d_in[0];
    const float* item_emb = (const float*)d_in[1];
    const float* hg_user  = (const float*)d_in[2];
    const float* hg_item  = (const float*)d_in[3];
    const float* adj_vals = (const float*)d_in[4];
    const float* W1       = (const float*)d_in[5];
    const float* b1       = (const float*)d_in[6];
    const float* W2       = (const float*)d_in[7];
    const float* b2       = (const float*)d_in[8];
    const int*   adj_rows = (const int*)d_in[9];
    const int*   adj_cols = (const int*)d_in[10];
    float* out = (float*)d_out;

    float* h1 = (float*)d_ws;                       // NUM_NODES x 64
    float* h2 = h1 + (size_t)NUM_NODES * D;         // NUM_NODES x 64

    // zero h1, h2
    long n4 = (long)(2 * (size_t)NUM_NODES * D / 4);
    int zb = (int)((n4 + 255) / 256);
    zero_kernel<<<zb, 256, 0, stream>>>((float4*)d_ws, n4);

    int E = in_sizes[4];
    int warps_per_block = 256 / 32;
    int spmm_blocks = (E + warps_per_block - 1) / warps_per_block;
    // layer 1: h1 = A * ego   (ego split user/item at M_USERS)
    spmm_kernel<<<spmm_blocks, 256, 0, stream>>>(user_emb, item_emb, M_USERS,
                                                 adj_rows, adj_cols, adj_vals, h1, E);
    // layer 2: h2 = A * h1    (contiguous: split = NUM_NODES)
    spmm_kernel<<<spmm_blocks, 256, 0, stream>>>(h1, h1, NUM_NODES,
                                                 adj_rows, adj_cols, adj_vals, h2, E);

    // fusion MLP + blend, 64 nodes per block (4 waves x 16 nodes)
    int fuse_blocks = (NUM_NODES + 63) / 64;
    fuse_kernel<<<fuse_blocks, 128, 0, stream>>>(user_emb, item_emb, hg_user, hg_item,
                                                 h1, h2, W1, b1, W2, b2, out);
}